// LightGCN_18382460027569
// MI455X (gfx1250) — compile-verified
//
#include <hip/hip_runtime.h>
#include <hip/hip_bf16.h>
#include <stdint.h>

// ---------------------------------------------------------------------------
// LightGCN propagation on MI455X (gfx1250).
// SpMM is scatter/atomic bound; cur+nxt embedding tables (2 x 76.8 MB) are
// L2-resident (192 MB L2). Edge streams are staged LDS-side with the Tensor
// Data Mover (tensor_load_to_lds, TENSORcnt), gathers use b128 loads, and the
// segment-sum uses native fp32 global atomics.
// ---------------------------------------------------------------------------

#define EMBED_DIM   64
#define TILE_EDGES  1024          // edges staged per workgroup via TDM
#define BLOCK       256

typedef __attribute__((ext_vector_type(4))) unsigned int u32x4;
typedef __attribute__((ext_vector_type(8))) int          i32x8;
typedef __attribute__((ext_vector_type(4))) int          i32x4;

// Issue one 1-D TDM tile load: global -> LDS.  lds_off is the byte offset of
// the destination inside the workgroup's LDS allocation (low 32 bits of the
// flat shared pointer).  n_valid elems are read; elems past tensor_dim0 are
// hardware zero-filled up to tile_elems.
__device__ __forceinline__ void tdm_load_1d(uint32_t lds_off, const void* gptr,
                                            uint32_t n_valid, uint32_t tile_elems) {
    uint64_t ga = (uint64_t)(uintptr_t)gptr;
    u32x4 g0;
    g0[0] = 1u;                                            // count=1 (valid user D#)
    g0[1] = lds_off;                                       // lds_addr (bytes)
    g0[2] = (uint32_t)ga;                                  // global_addr[31:0]
    g0[3] = (uint32_t)((ga >> 32) & 0x01FFFFFFu)           // global_addr[56:32]
            | (2u << 30);                                  // type = 2 ("image")
    i32x8 g1;
    g1[0] = (int)(2u << 16);                               // data_size = 4 bytes
    g1[1] = (int)((n_valid & 0xFFFFu) << 16);              // tensor_dim0[15:0]
    g1[2] = (int)(((n_valid >> 16) & 0xFFFFu)              // tensor_dim0[31:16]
            | (1u << 16));                                 // tensor_dim1 lo16 = 1
    g1[3] = (int)((tile_elems & 0xFFFFu) << 16);           // tensor_dim1 hi=0, tile_dim0
    g1[4] = 1;                                             // tile_dim1 = 1, tile_dim2 = 0
    g1[5] = (int)n_valid;                                  // tensor_dim0_stride lo (unused, 1 row)
    g1[6] = 0;
    g1[7] = 0;
    i32x4 z4; z4[0] = 0; z4[1] = 0; z4[2] = 0; z4[3] = 0;  // groups 2/3: <=2D tensor
    i32x8 z8;
    z8[0] = 0; z8[1] = 0; z8[2] = 0; z8[3] = 0;
    z8[4] = 0; z8[5] = 0; z8[6] = 0; z8[7] = 0;
    // 6-arg form (amdgpu-toolchain clang): (g0, g1, g2, g3, extra, cpol)
    __builtin_amdgcn_tensor_load_to_lds(g0, g1, z4, z4, z8, 0);
}

// cur = concat(user_emb, item_emb), float4-wide.
__global__ void k_concat(const float4* __restrict__ u, const float4* __restrict__ it,
                         float4* __restrict__ dst, int nu4, int ntot4) {
    int i = blockIdx.x * BLOCK + threadIdx.x;
    if (i >= ntot4) return;
    dst[i] = (i < nu4) ? u[i] : it[i - nu4];
}

__global__ void k_zero(float4* __restrict__ dst, int n4) {
    int i = blockIdx.x * BLOCK + threadIdx.x;
    if (i >= n4) return;
    float4 z; z.x = 0.f; z.y = 0.f; z.z = 0.f; z.w = 0.f;
    dst[i] = z;
}

// One 1024-edge tile per block.  Wave 0 DMAs rows/cols/vals into LDS via TDM,
// waits TENSORcnt, then 16 lanes per edge gather a float4 of cur[col] and
// atomically scatter-add v*x into nxt[row].
__global__ void __launch_bounds__(BLOCK)
k_spmm(const float* __restrict__ cur, float* __restrict__ nxt,
       const int* __restrict__ rows, const int* __restrict__ cols,
       const float* __restrict__ vals, int n_edges) {
    __shared__ int   s_rows[TILE_EDGES];
    __shared__ int   s_cols[TILE_EDGES];
    __shared__ float s_vals[TILE_EDGES];

    const int tile_start = blockIdx.x * TILE_EDGES;
    int remaining = n_edges - tile_start;
    if (remaining > TILE_EDGES) remaining = TILE_EDGES;
    if (remaining <= 0) return;

    if (threadIdx.x < 32u) {   // wave 0 only: TDM is a per-wave DMA op
        tdm_load_1d((uint32_t)(uintptr_t)s_rows, rows + tile_start,
                    (uint32_t)remaining, TILE_EDGES);
        tdm_load_1d((uint32_t)(uintptr_t)s_cols, cols + tile_start,
                    (uint32_t)remaining, TILE_EDGES);
        tdm_load_1d((uint32_t)(uintptr_t)s_vals, vals + tile_start,
                    (uint32_t)remaining, TILE_EDGES);
        __builtin_amdgcn_s_wait_tensorcnt(0);
    }
    __syncthreads();

    const int sub   = threadIdx.x & 15;   // which float4 of the 64-dim row
    const int eslot = threadIdx.x >> 4;   // 16 edges processed per pass

    #pragma unroll 4
    for (int p = 0; p < TILE_EDGES / 16; ++p) {
        const int e = p * 16 + eslot;
        if (e < remaining) {
            const int   r = s_rows[e];
            const int   c = s_cols[e];
            const float v = s_vals[e];
            const float4 x = *reinterpret_cast<const float4*>(
                cur + (size_t)c * EMBED_DIM + sub * 4);
            float* dst = nxt + (size_t)r * EMBED_DIM + sub * 4;
            __hip_atomic_fetch_add(dst + 0, v * x.x, __ATOMIC_RELAXED, __HIP_MEMORY_SCOPE_AGENT);
            __hip_atomic_fetch_add(dst + 1, v * x.y, __ATOMIC_RELAXED, __HIP_MEMORY_SCOPE_AGENT);
            __hip_atomic_fetch_add(dst + 2, v * x.z, __ATOMIC_RELAXED, __HIP_MEMORY_SCOPE_AGENT);
            __hip_atomic_fetch_add(dst + 3, v * x.w, __ATOMIC_RELAXED, __HIP_MEMORY_SCOPE_AGENT);
        }
    }
}

// out[j] (mode==0: =, mode==1: +=) buf[row(j)*64 + d], then * post_scale.
// j in [0, 2*batch*64): first batch*64 entries are user rows, rest item rows.
__global__ void k_gather_acc(const float* __restrict__ buf,
                             const int* __restrict__ users, const int* __restrict__ items,
                             int batch, int n_users, float* __restrict__ out,
                             int mode, float post_scale) {
    int j = blockIdx.x * BLOCK + threadIdx.x;
    const int total = 2 * batch * EMBED_DIM;
    if (j >= total) return;
    const int b = j >> 6;                 // row index in output
    const int d = j & 63;
    const int row = (b < batch) ? users[b] : (n_users + items[b - batch]);
    const float g = buf[(size_t)row * EMBED_DIM + d];
    float acc = (mode != 0) ? (out[j] + g) : g;
    out[j] = acc * post_scale;
}

extern "C" void kernel_launch(void* const* d_in, const int* in_sizes, int n_in,
                              void* d_out, int out_size, void* d_ws, size_t ws_size,
                              hipStream_t stream) {
    const float* user_emb = (const float*)d_in[0];
    const float* item_emb = (const float*)d_in[1];
    const float* adj_vals = (const float*)d_in[2];
    const int*   adj_rows = (const int*)d_in[3];
    const int*   adj_cols = (const int*)d_in[4];
    const int*   users    = (const int*)d_in[5];
    const int*   items    = (const int*)d_in[6];

    const int n_users = in_sizes[0] / EMBED_DIM;
    const int n_items = in_sizes[1] / EMBED_DIM;
    const int n_edges = in_sizes[2];
    const int batch   = in_sizes[5];
    const int n_total = n_users + n_items;

    float* cur = (float*)d_ws;
    float* nxt = cur + (size_t)n_total * EMBED_DIM;
    float* out = (float*)d_out;

    const int ntot4 = n_total * (EMBED_DIM / 4);
    const int nu4   = n_users * (EMBED_DIM / 4);
    const int g_tab = (ntot4 + BLOCK - 1) / BLOCK;

    // e^0 = concat(user_emb, item_emb)
    k_concat<<<g_tab, BLOCK, 0, stream>>>((const float4*)user_emb,
                                          (const float4*)item_emb,
                                          (float4*)cur, nu4, ntot4);

    const int out_elems = 2 * batch * EMBED_DIM;
    const int g_out = (out_elems + BLOCK - 1) / BLOCK;
    // acc := e^0 (gathered at batch rows)
    k_gather_acc<<<g_out, BLOCK, 0, stream>>>(cur, users, items, batch, n_users,
                                              out, /*mode=*/0, 1.0f);

    const int n_tiles = (n_edges + TILE_EDGES - 1) / TILE_EDGES;
    float* a = cur;
    float* b = nxt;
    for (int layer = 0; layer < 3; ++layer) {
        k_zero<<<g_tab, BLOCK, 0, stream>>>((float4*)b, ntot4);
        k_spmm<<<n_tiles, BLOCK, 0, stream>>>(a, b, adj_rows, adj_cols, adj_vals, n_edges);
        const float ps = (layer == 2) ? 0.25f : 1.0f;   // mean over 4 layer embeddings
        k_gather_acc<<<g_out, BLOCK, 0, stream>>>(b, users, items, batch, n_users,
                                                  out, /*mode=*/1, ps);
        float* t = a; a = b; b = t;
    }
}